// TinyMixedHeteroLinkPredictor_40776419508772
// MI455X (gfx1250) — compile-verified
//
#include <hip/hip_runtime.h>

typedef __attribute__((ext_vector_type(2))) float  vfloat2;
typedef __attribute__((ext_vector_type(4))) float  vfloat4;
typedef __attribute__((ext_vector_type(8))) float  vfloat8;
typedef __attribute__((ext_vector_type(4))) int    vint4;

// ---------------------------------------------------------------------------
// Phase 1: per-node scalar scores via V_WMMA_F32_16X16X4_F32.
// D = A(16x4) x B(4x16):
//   A rows: row0 = u_src (Ws[0:4] folded through W), row1 = u_dst, rows 2..15 = 0
//   B cols: 16 nodes' 4-float feature vectors
//   D VGPR0 lanes 0..15 = src-score of node(lane); VGPR1 = dst-score.
// One WMMA for the author table, one for the paper table, per 16-node tile.
// ---------------------------------------------------------------------------
__global__ __launch_bounds__(256) void node_score_kernel(
    const float* __restrict__ author_x, const float* __restrict__ paper_x,
    const float* __restrict__ Wa, const float* __restrict__ ba,
    const float* __restrict__ Wp, const float* __restrict__ bp,
    const float* __restrict__ Ws,
    float* __restrict__ S_src, float* __restrict__ S_dst, int N)
{
    const int lane = threadIdx.x & 31;
    const int waveId = blockIdx.x * 8 + (threadIdx.x >> 5);
    const int tileBase = waveId * 16;
    if (tileBase >= N) return;                 // uniform per wave

    const int row  = lane & 15;                // M (A) / N column (B)
    const int half = lane >> 4;
    const int ka   = half * 2;                 // this lane's K pair: ka, ka+1

    // A operands: combined weight vectors (rows 0/1), zeros elsewhere.
    vfloat2 aA = {0.f, 0.f}, aP = {0.f, 0.f};
    if (row < 2) {
        const float* w = Ws + row * 4;         // row0 -> Ws[0:4], row1 -> Ws[4:8]
        float s0 = 0.f, s1 = 0.f, p0 = 0.f, p1 = 0.f;
#pragma unroll
        for (int j = 0; j < 4; ++j) {
            s0 += w[j] * Wa[j * 4 + ka];
            s1 += w[j] * Wa[j * 4 + ka + 1];
            p0 += w[j] * Wp[j * 4 + ka];
            p1 += w[j] * Wp[j * 4 + ka + 1];
        }
        aA.x = s0; aA.y = s1; aP.x = p0; aP.y = p1;
    }

    // Folded bias constants (uniform across wave).
    float c_a_src = 0.f, c_a_dst = 0.f, c_p_src = 0.f, c_p_dst = 0.f;
#pragma unroll
    for (int j = 0; j < 4; ++j) {
        c_a_src += Ws[j]     * ba[j];
        c_a_dst += Ws[4 + j] * ba[j];
        c_p_src += Ws[j]     * bp[j];
        c_p_dst += Ws[4 + j] * bp[j];
    }

    // B operands: lane holds features (ka, ka+1) of node tileBase+row.
    const int node  = tileBase + row;
    const int nodeC = node < N ? node : N - 1; // clamp tail, keep EXEC all-1s
    vfloat2 bA = *(const vfloat2*)(author_x + (size_t)nodeC * 4 + ka);
    vfloat2 bP = *(const vfloat2*)(paper_x  + (size_t)nodeC * 4 + ka);

    vfloat8 cz = {};
    vfloat8 dA = __builtin_amdgcn_wmma_f32_16x16x4_f32(
        false, aA, false, bA, (short)0, cz, false, false);
    vfloat8 dP = __builtin_amdgcn_wmma_f32_16x16x4_f32(
        false, aP, false, bP, (short)0, cz, false, false);

    if (half == 0 && node < N) {
        // lanes 0..15: D[0] = (M=0,N=lane) = src score, D[1] = dst score
        S_src[node]     = dA[0] + c_a_src;     // author-as-src
        S_dst[node]     = dA[1] + c_a_dst;     // author-as-dst
        S_src[N + node] = dP[0] + c_p_src;     // paper-as-src
        S_dst[N + node] = dP[1] + c_p_dst;     // paper-as-dst
    }
}

// ---------------------------------------------------------------------------
// Phase 2: per-edge 4B gathers from the L2-resident score tables.
// 4 edges per thread: 3x b128 streaming loads, 8 scalar gathers, 1 b128 store.
// ---------------------------------------------------------------------------
__global__ __launch_bounds__(256) void edge_score_kernel(
    const int* __restrict__ src_index, const int* __restrict__ dst_index,
    const int* __restrict__ edge_type,
    const int* __restrict__ esp, const int* __restrict__ edp,
    const float* __restrict__ bs,
    const float* __restrict__ S_src, const float* __restrict__ S_dst,
    float* __restrict__ out, int E, int N)
{
    const int mask_s = (esp[0] & 1) | ((esp[1] & 1) << 1) |
                       ((esp[2] & 1) << 2) | ((esp[3] & 1) << 3);
    const int mask_d = (edp[0] & 1) | ((edp[1] & 1) << 1) |
                       ((edp[2] & 1) << 2) | ((edp[3] & 1) << 3);
    const float b0 = bs[0];

    const long long i0 = (long long)(blockIdx.x * blockDim.x + threadIdx.x) * 4;
    if (i0 + 3 < (long long)E) {
        vint4 s = __builtin_nontemporal_load((const vint4*)(src_index + i0));
        vint4 d = __builtin_nontemporal_load((const vint4*)(dst_index + i0));
        vint4 t = __builtin_nontemporal_load((const vint4*)(edge_type + i0));
        vfloat4 r;
        r.x = S_src[((mask_s >> t.x) & 1) * N + s.x] +
              S_dst[((mask_d >> t.x) & 1) * N + d.x] + b0;
        r.y = S_src[((mask_s >> t.y) & 1) * N + s.y] +
              S_dst[((mask_d >> t.y) & 1) * N + d.y] + b0;
        r.z = S_src[((mask_s >> t.z) & 1) * N + s.z] +
              S_dst[((mask_d >> t.z) & 1) * N + d.z] + b0;
        r.w = S_src[((mask_s >> t.w) & 1) * N + s.w] +
              S_dst[((mask_d >> t.w) & 1) * N + d.w] + b0;
        __builtin_nontemporal_store(r, (vfloat4*)(out + i0));
    } else {
        for (long long i = i0; i < (long long)E; ++i) {
            int t = edge_type[i];
            out[i] = S_src[((mask_s >> t) & 1) * N + src_index[i]] +
                     S_dst[((mask_d >> t) & 1) * N + dst_index[i]] + b0;
        }
    }
}

// ---------------------------------------------------------------------------
// Fallback (ws too small): fused per-edge compute, 16B gathers of raw features.
// ---------------------------------------------------------------------------
__global__ __launch_bounds__(256) void fused_edge_kernel(
    const float* __restrict__ author_x, const float* __restrict__ paper_x,
    const int* __restrict__ src_index, const int* __restrict__ dst_index,
    const int* __restrict__ edge_type,
    const int* __restrict__ esp, const int* __restrict__ edp,
    const float* __restrict__ Wa, const float* __restrict__ ba,
    const float* __restrict__ Wp, const float* __restrict__ bp,
    const float* __restrict__ Ws, const float* __restrict__ bs,
    float* __restrict__ out, int E)
{
    // u[table][endpoint][k], c[table][endpoint]
    float ua_s[4], ua_d[4], up_s[4], up_d[4];
    float ca_s = 0.f, ca_d = 0.f, cp_s = 0.f, cp_d = 0.f;
#pragma unroll
    for (int k = 0; k < 4; ++k) {
        float as = 0.f, ad = 0.f, ps = 0.f, pd = 0.f;
#pragma unroll
        for (int j = 0; j < 4; ++j) {
            as += Ws[j]     * Wa[j * 4 + k];
            ad += Ws[4 + j] * Wa[j * 4 + k];
            ps += Ws[j]     * Wp[j * 4 + k];
            pd += Ws[4 + j] * Wp[j * 4 + k];
        }
        ua_s[k] = as; ua_d[k] = ad; up_s[k] = ps; up_d[k] = pd;
    }
#pragma unroll
    for (int j = 0; j < 4; ++j) {
        ca_s += Ws[j] * ba[j];     ca_d += Ws[4 + j] * ba[j];
        cp_s += Ws[j] * bp[j];     cp_d += Ws[4 + j] * bp[j];
    }
    const int mask_s = (esp[0] & 1) | ((esp[1] & 1) << 1) |
                       ((esp[2] & 1) << 2) | ((esp[3] & 1) << 3);
    const int mask_d = (edp[0] & 1) | ((edp[1] & 1) << 1) |
                       ((edp[2] & 1) << 2) | ((edp[3] & 1) << 3);
    const float b0 = bs[0];

    const long long i = blockIdx.x * (long long)blockDim.x + threadIdx.x;
    if (i >= (long long)E) return;
    const int t  = edge_type[i];
    const int sp = (mask_s >> t) & 1;
    const int dp = (mask_d >> t) & 1;
    vfloat4 xs = *(const vfloat4*)((sp ? paper_x : author_x) + (size_t)src_index[i] * 4);
    vfloat4 xd = *(const vfloat4*)((dp ? paper_x : author_x) + (size_t)dst_index[i] * 4);
    float r = (sp ? cp_s : ca_s) + (dp ? cp_d : ca_d) + b0;
#pragma unroll
    for (int k = 0; k < 4; ++k) {
        r += xs[k] * (sp ? up_s[k] : ua_s[k]);
        r += xd[k] * (dp ? up_d[k] : ua_d[k]);
    }
    out[i] = r;
}

extern "C" void kernel_launch(void* const* d_in, const int* in_sizes, int n_in,
                              void* d_out, int out_size, void* d_ws, size_t ws_size,
                              hipStream_t stream) {
    const float* author_x = (const float*)d_in[0];
    const float* paper_x  = (const float*)d_in[1];
    const int*   src_idx  = (const int*)d_in[2];
    const int*   dst_idx  = (const int*)d_in[3];
    const int*   etype    = (const int*)d_in[4];
    const int*   esp      = (const int*)d_in[5];
    const int*   edp      = (const int*)d_in[6];
    const float* Wa       = (const float*)d_in[7];
    const float* ba       = (const float*)d_in[8];
    const float* Wp       = (const float*)d_in[9];
    const float* bp       = (const float*)d_in[10];
    const float* Ws       = (const float*)d_in[11];
    const float* bs       = (const float*)d_in[12];
    float* out = (float*)d_out;

    const int N = in_sizes[0] / 4;   // 2,000,000
    const int E = in_sizes[2];       // 16,777,216

    const size_t wsNeeded = (size_t)4 * (size_t)N * sizeof(float);
    if (ws_size >= wsNeeded) {
        float* S_src = (float*)d_ws;              // [2][N]: author, paper as src
        float* S_dst = S_src + (size_t)2 * N;     // [2][N]: author, paper as dst

        const int tiles   = (N + 15) / 16;
        const int blocks1 = (tiles + 7) / 8;      // 8 waves (256 thr) per block
        node_score_kernel<<<blocks1, 256, 0, stream>>>(
            author_x, paper_x, Wa, ba, Wp, bp, Ws, S_src, S_dst, N);

        const long long thr2    = ((long long)E + 3) / 4;
        const int       blocks2 = (int)((thr2 + 255) / 256);
        edge_score_kernel<<<blocks2, 256, 0, stream>>>(
            src_idx, dst_idx, etype, esp, edp, bs, S_src, S_dst, out, E, N);
    } else {
        const int blocks = (int)(((long long)E + 255) / 256);
        fused_edge_kernel<<<blocks, 256, 0, stream>>>(
            author_x, paper_x, src_idx, dst_idx, etype, esp, edp,
            Wa, ba, Wp, bp, Ws, bs, out, E);
    }
}